// Feat_init_32598801777024
// MI455X (gfx1250) — compile-verified
//
#include <hip/hip_runtime.h>
#include <hip/hip_bf16.h>

#define DIMX   256
#define HEADSX 8
#define DHX    32
#define NPADX  16
#define BSZX   2048
#define LX     128
#define NORGX  229376
#define NTOTX  262144
#define EORGX  458752
#define ESELFX 65536
#define EPADX  262144
#define ETOTX  786432

typedef __attribute__((ext_vector_type(2))) float v2f;
typedef __attribute__((ext_vector_type(8))) float v8f;

static __device__ __forceinline__ v8f wmma4(v2f a, v2f b, v8f c) {
    // D(16x16,f32) = A(16x4,f32) * B(4x16,f32) + C
    return __builtin_amdgcn_wmma_f32_16x16x4_f32(false, a, false, b, (short)0, c,
                                                 false, false);
}

// ---------------- q projection (batch-invariant): q = (Qemb@Wq + bq)/sqrt(dh) ----
__global__ void k_qproj(const float* __restrict__ Qemb, const float* __restrict__ Wq,
                        const float* __restrict__ bq, float* __restrict__ qout) {
    int qi = blockIdx.x, d = threadIdx.x;
    float acc = 0.f;
    for (int k = 0; k < DIMX; ++k) acc += Qemb[qi * DIMX + k] * Wq[k * DIMX + d];
    qout[qi * DIMX + d] = (acc + bq[d]) * 0.17677669529663687f; // 1/sqrt(32)
}

// ---------------- atom encoder: node_feat[n] = sum_f atom_emb[f, x[n,f]] ---------
__global__ void k_atom(const int* __restrict__ x, const float* __restrict__ atom_emb,
                       float* __restrict__ node_out) {
    int n = blockIdx.x, d = threadIdx.x;
    __shared__ int xi[9];
    if (d < 9) xi[d] = x[n * 9 + d];
    __syncthreads();
    float s = 0.f;
#pragma unroll
    for (int f = 0; f < 9; ++f) s += atom_emb[(f * 128 + xi[f]) * DIMX + d];
    node_out[(size_t)n * DIMX + d] = s;
}

// ---------------- bond encoder: org edges -----------------------------------------
__global__ void k_bond(const int* __restrict__ ea, const float* __restrict__ bond_emb,
                       float* __restrict__ edge_out) {
    int e = blockIdx.x, d = threadIdx.x;
    __shared__ int bi[3];
    if (d < 3) bi[d] = ea[e * 3 + d];
    __syncthreads();
    float s = 0.f;
#pragma unroll
    for (int f = 0; f < 3; ++f) s += bond_emb[(f * 32 + bi[f]) * DIMX + d];
    edge_out[(size_t)e * DIMX + d] = s;
}

// ---------------- self-loop edges --------------------------------------------------
__global__ void k_self(const float* __restrict__ selfloop, float* __restrict__ edge_out) {
    size_t e = (size_t)EORGX + blockIdx.x;
    edge_out[e * DIMX + threadIdx.x] = selfloop[threadIdx.x];
}

// ---------------- fused cross-attention (one workgroup per batch element) ----------
#define KVSTR 36
#define SSTR  132
#define QSTR  260

__global__ __launch_bounds__(256) void k_attn(
    const float* __restrict__ memory, const unsigned char* __restrict__ cross_mask,
    const float* __restrict__ qscaled,
    const float* __restrict__ Wk, const float* __restrict__ bk,
    const float* __restrict__ Wv, const float* __restrict__ bv,
    const float* __restrict__ Wo, const float* __restrict__ bo,
    float* __restrict__ node_out) {
    __shared__ float qbuf[NPADX * QSTR];   // scaled q, [16][260]
    __shared__ float ctxbuf[NPADX * QSTR]; // ctx,       [16][260]
    __shared__ float kvbuf[LX * KVSTR];    // K_h then V_h, [128][36]
    __shared__ float sbuf[NPADX * SSTR];   // scores/attn,  [16][132]

    const int tid = threadIdx.x;
    const int laneLo = tid & 15;
    const int laneHi = (tid >> 4) & 1;
    const int wv = tid >> 5; // wave id 0..7
    const int b = blockIdx.x;
    const float* memb = memory + (size_t)b * LX * DIMX;
    const unsigned char* maskb = cross_mask + (size_t)b * NPADX * LX;

    for (int u = tid; u < NPADX * DIMX; u += 256)
        qbuf[(u >> 8) * QSTR + (u & 255)] = qscaled[u];
    __syncthreads();

    for (int h = 0; h < HEADSX; ++h) {
        // ---- K_h = memb @ Wk[:, h*32:+32] + bk -> kvbuf[key][dh]
        // wave w: tileM = w, tileN in {0,1}; A shared across both tiles.
        {
            v8f acc0 = {}, acc1 = {};
            const float* arow  = memb + (wv * 16 + laneLo) * DIMX + 2 * laneHi;
            const float* bcol0 = Wk + h * DHX + laneLo;
            const float* bcol1 = Wk + h * DHX + 16 + laneLo;
            for (int kk = 0; kk < DIMX; kk += 4) {
                v2f a = *(const v2f*)(arow + kk);
                int r0 = (kk + 2 * laneHi) * DIMX, r1 = (kk + 1 + 2 * laneHi) * DIMX;
                v2f b0, b1;
                b0.x = bcol0[r0]; b0.y = bcol0[r1];
                b1.x = bcol1[r0]; b1.y = bcol1[r1];
                acc0 = wmma4(a, b0, acc0);
                acc1 = wmma4(a, b1, acc1);
            }
            float bi0 = bk[h * DHX + laneLo];
            float bi1 = bk[h * DHX + 16 + laneLo];
#pragma unroll
            for (int v = 0; v < 8; ++v) {
                int key = wv * 16 + v + 8 * laneHi;
                kvbuf[key * KVSTR + laneLo]      = acc0[v] + bi0;
                kvbuf[key * KVSTR + 16 + laneLo] = acc1[v] + bi1;
            }
        }
        __syncthreads();

        // ---- S_h = q_h @ K_h^T (scale pre-folded), mask, -> sbuf[q][key]
        {
            int tileN = wv; // key tile 0..7
            v8f acc = {};
            const float* arow = qbuf + laneLo * QSTR + h * DHX + 2 * laneHi;
            const float* brow = kvbuf + (tileN * 16 + laneLo) * KVSTR + 2 * laneHi;
#pragma unroll
            for (int kk = 0; kk < DHX; kk += 4) {
                v2f a = *(const v2f*)(arow + kk);
                v2f bb = *(const v2f*)(brow + kk);
                acc = wmma4(a, bb, acc);
            }
            int key = tileN * 16 + laneLo;
#pragma unroll
            for (int v = 0; v < 8; ++v) {
                int q = v + 8 * laneHi;
                float val = acc[v];
                if (maskb[q * LX + key]) val = -1e9f;
                sbuf[q * SSTR + key] = val;
            }
        }
        __syncthreads();

        // ---- softmax over 128 keys (one thread per query row)
        if (tid < NPADX) {
            float m = -3.4e38f;
            for (int k = 0; k < LX; ++k) m = fmaxf(m, sbuf[tid * SSTR + k]);
            float s = 0.f;
            for (int k = 0; k < LX; ++k) {
                float e = __expf(sbuf[tid * SSTR + k] - m);
                sbuf[tid * SSTR + k] = e;
                s += e;
            }
            float inv = 1.f / s;
            for (int k = 0; k < LX; ++k) sbuf[tid * SSTR + k] *= inv;
        }
        __syncthreads();

        // ---- V_h = memb @ Wv[:, h*32:+32] + bv -> kvbuf (overwrites K_h)
        {
            v8f acc0 = {}, acc1 = {};
            const float* arow  = memb + (wv * 16 + laneLo) * DIMX + 2 * laneHi;
            const float* bcol0 = Wv + h * DHX + laneLo;
            const float* bcol1 = Wv + h * DHX + 16 + laneLo;
            for (int kk = 0; kk < DIMX; kk += 4) {
                v2f a = *(const v2f*)(arow + kk);
                int r0 = (kk + 2 * laneHi) * DIMX, r1 = (kk + 1 + 2 * laneHi) * DIMX;
                v2f b0, b1;
                b0.x = bcol0[r0]; b0.y = bcol0[r1];
                b1.x = bcol1[r0]; b1.y = bcol1[r1];
                acc0 = wmma4(a, b0, acc0);
                acc1 = wmma4(a, b1, acc1);
            }
            float bi0 = bv[h * DHX + laneLo];
            float bi1 = bv[h * DHX + 16 + laneLo];
#pragma unroll
            for (int v = 0; v < 8; ++v) {
                int key = wv * 16 + v + 8 * laneHi;
                kvbuf[key * KVSTR + laneLo]      = acc0[v] + bi0;
                kvbuf[key * KVSTR + 16 + laneLo] = acc1[v] + bi1;
            }
        }
        __syncthreads();

        // ---- ctx_h = attn @ V_h  (16x32, two 16x16 tiles, waves 0-1)
        if (wv < 2) {
            int tileN = wv;
            v8f acc = {};
            const float* arow = sbuf + laneLo * SSTR + 2 * laneHi;
#pragma unroll 8
            for (int kk = 0; kk < LX; kk += 4) {
                v2f a = *(const v2f*)(arow + kk);
                v2f bb;
                bb.x = kvbuf[(kk + 2 * laneHi) * KVSTR + tileN * 16 + laneLo];
                bb.y = kvbuf[(kk + 1 + 2 * laneHi) * KVSTR + tileN * 16 + laneLo];
                acc = wmma4(a, bb, acc);
            }
#pragma unroll
            for (int v = 0; v < 8; ++v) {
                int q = v + 8 * laneHi;
                ctxbuf[q * QSTR + h * DHX + tileN * 16 + laneLo] = acc[v];
            }
        }
        __syncthreads();
    }

    // ---- out = ctx @ Wo + bo -> node_feat rows [NORG + b*16 .. +16)
    // wave w: tileN = w and w+8 fused (A from LDS shared).
    {
        v8f acc0 = {}, acc1 = {};
        const float* arow  = ctxbuf + laneLo * QSTR + 2 * laneHi;
        const float* bcol0 = Wo + wv * 16 + laneLo;
        const float* bcol1 = Wo + (wv + 8) * 16 + laneLo;
        for (int kk = 0; kk < DIMX; kk += 4) {
            v2f a = *(const v2f*)(arow + kk);
            int r0 = (kk + 2 * laneHi) * DIMX, r1 = (kk + 1 + 2 * laneHi) * DIMX;
            v2f b0, b1;
            b0.x = bcol0[r0]; b0.y = bcol0[r1];
            b1.x = bcol1[r0]; b1.y = bcol1[r1];
            acc0 = wmma4(a, b0, acc0);
            acc1 = wmma4(a, b1, acc1);
        }
        float bo0 = bo[wv * 16 + laneLo];
        float bo1 = bo[(wv + 8) * 16 + laneLo];
#pragma unroll
        for (int v = 0; v < 8; ++v) {
            int q = v + 8 * laneHi;
            size_t node = (size_t)NORGX + (size_t)b * NPADX + q;
            node_out[node * DIMX + wv * 16 + laneLo]       = acc0[v] + bo0;
            node_out[node * DIMX + (wv + 8) * 16 + laneLo] = acc1[v] + bo1;
        }
    }
}

// ---------------- pad-edge MLP: relu(concat(nf[i],nf[j])) @ Wl + bl ---------------
#define ROWS 16
#define ASTR 516

__global__ __launch_bounds__(256) void k_padedge(
    const int* __restrict__ edge_index, const float* __restrict__ node_feat,
    const float* __restrict__ Wl, const float* __restrict__ bl,
    float* __restrict__ edge_out) {
    __shared__ float Ast[ROWS * ASTR]; // 16 x 512 relu'd gathered rows (padded)
    const int tid = threadIdx.x;
    const int laneLo = tid & 15;
    const int laneHi = (tid >> 4) & 1;
    const int wv = tid >> 5;
    const long long P0 = (long long)EORGX + ESELFX;
    const long long e0 = (long long)blockIdx.x * ROWS;

    // stage A: rows = edges, cols = relu(node_feat[i]) || relu(node_feat[j])
    for (int u = tid; u < ROWS * 128; u += 256) {
        int r = u >> 7, c4 = u & 127;
        int src, col;
        if (c4 < 64) { src = edge_index[P0 + e0 + r];                    col = c4 * 4; }
        else         { src = edge_index[(long long)ETOTX + P0 + e0 + r]; col = (c4 - 64) * 4; }
        float4 vsrc = *(const float4*)(node_feat + (size_t)src * DIMX + col);
        float* dst = Ast + r * ASTR + c4 * 4;
        dst[0] = fmaxf(vsrc.x, 0.f);
        dst[1] = fmaxf(vsrc.y, 0.f);
        dst[2] = fmaxf(vsrc.z, 0.f);
        dst[3] = fmaxf(vsrc.w, 0.f);
    }
    __syncthreads();

    // wave w: tileN = w and w+8 fused; A (staged LDS tile) shared.
    {
        v8f acc0 = {}, acc1 = {};
        const float* arow  = Ast + laneLo * ASTR + 2 * laneHi;
        const float* bcol0 = Wl + wv * 16 + laneLo;
        const float* bcol1 = Wl + (wv + 8) * 16 + laneLo;
        for (int kk = 0; kk < 2 * DIMX; kk += 4) {
            v2f a = *(const v2f*)(arow + kk);
            int r0 = (kk + 2 * laneHi) * DIMX, r1 = (kk + 1 + 2 * laneHi) * DIMX;
            v2f b0, b1;
            b0.x = bcol0[r0]; b0.y = bcol0[r1];
            b1.x = bcol1[r0]; b1.y = bcol1[r1];
            acc0 = wmma4(a, b0, acc0);
            acc1 = wmma4(a, b1, acc1);
        }
        float bl0 = bl[wv * 16 + laneLo];
        float bl1 = bl[(wv + 8) * 16 + laneLo];
#pragma unroll
        for (int v = 0; v < 8; ++v) {
            int m = v + 8 * laneHi;
            size_t row = (size_t)(P0 + e0) + m;
            edge_out[row * DIMX + wv * 16 + laneLo]       = acc0[v] + bl0;
            edge_out[row * DIMX + (wv + 8) * 16 + laneLo] = acc1[v] + bl1;
        }
    }
}

extern "C" void kernel_launch(void* const* d_in, const int* in_sizes, int n_in,
                              void* d_out, int out_size, void* d_ws, size_t ws_size,
                              hipStream_t stream) {
    const int* x            = (const int*)d_in[0];
    const int* edge_attr    = (const int*)d_in[1];
    const int* edge_index   = (const int*)d_in[2];
    const float* memory     = (const float*)d_in[8];
    const unsigned char* cm = (const unsigned char*)d_in[9];
    const float* Qemb       = (const float*)d_in[10];
    const float* atom_emb   = (const float*)d_in[11];
    const float* bond_emb   = (const float*)d_in[12];
    const float* selfloop   = (const float*)d_in[13];
    const float* Wq = (const float*)d_in[14]; const float* bq = (const float*)d_in[15];
    const float* Wk = (const float*)d_in[16]; const float* bk = (const float*)d_in[17];
    const float* Wv = (const float*)d_in[18]; const float* bv = (const float*)d_in[19];
    const float* Wo = (const float*)d_in[20]; const float* bo = (const float*)d_in[21];
    const float* Wl = (const float*)d_in[22]; const float* bl = (const float*)d_in[23];

    float* node_out = (float*)d_out;
    float* edge_out = node_out + (size_t)NTOTX * DIMX;
    float* qscaled  = (float*)d_ws; // 16*256 floats

    k_qproj<<<NPADX, DIMX, 0, stream>>>(Qemb, Wq, bq, qscaled);
    k_atom<<<NORGX, DIMX, 0, stream>>>(x, atom_emb, node_out);
    k_bond<<<EORGX, DIMX, 0, stream>>>(edge_attr, bond_emb, edge_out);
    k_self<<<ESELFX, DIMX, 0, stream>>>(selfloop, edge_out);
    k_attn<<<BSZX, 256, 0, stream>>>(memory, cm, qscaled, Wk, bk, Wv, bv, Wo, bo, node_out);
    k_padedge<<<EPADX / ROWS, 256, 0, stream>>>(edge_index, node_out, Wl, bl, edge_out);
}